// RigidTransformation3DImputation_19585050870113
// MI455X (gfx1250) — compile-verified
//
#include <hip/hip_runtime.h>

// ============================================================================
// RigidTransformation3DImputation on MI455X (gfx1250, wave32, WMMA)
//
//  K0 twiddle      : precompute shifted-DFT and inverse-DFT 64x64 matrices
//  K1 theta_params : per-batch rotation M = (Rz Ry Rz)^T and translation c
//  K2 warp         : trilinear warps -> M1_t, M2_t (to d_out) and X_t (ws)
//  K3 plane fwd    : per (b,d) plane, shifted-DFT along W then H via
//                    V_WMMA_F32_16X16X4_F32 complex GEMMs in LDS -> FX, FY
//  K4 d-line       : fwd-DFT along D for FX & FY, mask MAC, inverse-DFT
//                    along D -> S1 (fused, one LDS pass)
//  K5 plane inv    : inverse-DFT along W then H, emit real part -> d_out
//
// fftshift∘fft∘fftshift per axis == DFT with F[k,j] = (-1)^(k+j) e^{-2πi kj/64}
// (sign baked into twiddles; no data rolls). ifftn uses (1/64) e^{+2πi kj/64}
// per axis. Twiddle tables (64 KB) stay L2-resident (192 MB L2).
// Global->LDS staging uses CDNA5 async-to-LDS (ASYNCcnt) when available.
// ============================================================================

typedef __attribute__((ext_vector_type(2))) float v2f;
typedef __attribute__((ext_vector_type(8))) float v8f;
typedef int vs4i __attribute__((vector_size(16)));   // matches async builtin pointee

#define NVOX 262144              // 64^3
#define NB   16
#define NTOT (NB * NVOX)         // 4194304
#define PI_F 3.14159265358979323846f

// ---------------------------------------------------------------------------
// Global -> LDS staging: async-copy path (gfx1250) with vectorized fallback
// ---------------------------------------------------------------------------
#if defined(__AMDGCN__) && __has_builtin(__builtin_amdgcn_global_load_async_to_lds_b128)
#define USE_ASYNC_LDS 1
#else
#define USE_ASYNC_LDS 0
#endif

__device__ __forceinline__ void copy_g2lds_b128(void* lds, const void* g)
{
#if USE_ASYNC_LDS
  __builtin_amdgcn_global_load_async_to_lds_b128(
      (__attribute__((address_space(1))) vs4i*)(unsigned long long)g,
      (__attribute__((address_space(3))) vs4i*)(unsigned)(unsigned long long)lds,
      0, 0);
#else
  *(float4*)lds = *(const float4*)g;
#endif
}

__device__ __forceinline__ void async_join()
{
#if USE_ASYNC_LDS
#if __has_builtin(__builtin_amdgcn_s_wait_asynccnt)
  __builtin_amdgcn_s_wait_asynccnt(0);
#else
  asm volatile("s_wait_asynccnt 0x0" ::: "memory");
#endif
#endif
  __syncthreads();
}

// ---------------------------------------------------------------------------
// Complex 16x16 tile GEMM over K=64 using V_WMMA_F32_16X16X4_F32.
// A is 16x64 row-major (contiguous K, row stride lda).
// B is 64x16 with generic strides: elem(k,n) = B[k*ldbr + n*ldbc].
// AREAL:  A has zero imaginary part.  REONLY: only real part of C needed.
// ---------------------------------------------------------------------------
template<bool AREAL, bool REONLY>
__device__ __forceinline__ void cgemm_tile(
    const float* __restrict__ Are, const float* __restrict__ Aim, int lda,
    const float* __restrict__ Bre, const float* __restrict__ Bim, int ldbr, int ldbc,
    int lane, v8f& outRe, v8f& outIm)
{
  const int half = (lane >> 4) & 1;
  const int lr   = lane & 15;
  v8f acc_rr = {}; v8f acc_ii = {}; v8f acc_ri = {}; v8f acc_ir = {};
  #pragma unroll
  for (int k0 = 0; k0 < 64; k0 += 4) {
    const int ka = k0 + 2 * half;       // A: v0 holds K=ka, v1 holds K=ka+1
    v2f ar, br, bi;
    ar.x = Are[lr * lda + ka];
    ar.y = Are[lr * lda + ka + 1];
    br.x = Bre[ka * ldbr + lr * ldbc];
    br.y = Bre[(ka + 1) * ldbr + lr * ldbc];
    bi.x = Bim[ka * ldbr + lr * ldbc];
    bi.y = Bim[(ka + 1) * ldbr + lr * ldbc];
    acc_rr = __builtin_amdgcn_wmma_f32_16x16x4_f32(false, ar, false, br, (short)0, acc_rr, false, false);
    if constexpr (!REONLY)
      acc_ri = __builtin_amdgcn_wmma_f32_16x16x4_f32(false, ar, false, bi, (short)0, acc_ri, false, false);
    if constexpr (!AREAL) {
      v2f ai;
      ai.x = Aim[lr * lda + ka];
      ai.y = Aim[lr * lda + ka + 1];
      acc_ii = __builtin_amdgcn_wmma_f32_16x16x4_f32(false, ai, false, bi, (short)0, acc_ii, false, false);
      if constexpr (!REONLY)
        acc_ir = __builtin_amdgcn_wmma_f32_16x16x4_f32(false, ai, false, br, (short)0, acc_ir, false, false);
    }
  }
  outRe = AREAL ? acc_rr : (acc_rr - acc_ii);
  if constexpr (!REONLY) outIm = AREAL ? acc_ri : (acc_ri + acc_ir);
  else                   outIm = v8f{};
}

// C/D layout: VGPR r holds (M = r + 8*half, N = lane&15)
__device__ __forceinline__ void store_tile(float* __restrict__ dst, int ld,
                                           const v8f& v, int lane)
{
  const int half = (lane >> 4) & 1;
  const int lr   = lane & 15;
  #pragma unroll
  for (int r = 0; r < 8; ++r) dst[(r + 8 * half) * ld + lr] = v[r];
}

// ---------------------------------------------------------------------------
// K0: twiddle tables.  TF = (-1)^(k+j) e^{-2πi kj/64},  TI = (1/64) e^{+2πi kj/64}
// ---------------------------------------------------------------------------
__global__ __launch_bounds__(256) void twiddle_kernel(
    float* __restrict__ TFre, float* __restrict__ TFim,
    float* __restrict__ TIre, float* __restrict__ TIim)
{
  const int tid = threadIdx.x;
  #pragma unroll
  for (int i = 0; i < 16; ++i) {
    const int idx = tid + 256 * i;
    const int k = idx >> 6, j = idx & 63;
    const int pr = (k * j) & 63;
    float s, c;
    sincosf(-(PI_F / 32.f) * (float)pr, &s, &c);
    const float sg = ((k + j) & 1) ? -1.f : 1.f;
    TFre[idx] = sg * c; TFim[idx] = sg * s;
    sincosf((PI_F / 32.f) * (float)pr, &s, &c);
    TIre[idx] = c * (1.f / 64.f); TIim[idx] = s * (1.f / 64.f);
  }
}

// ---------------------------------------------------------------------------
// K1: per-batch affine params. M = (Rz(phi)Ry(th)Rz(psi))^T, c = -M*loc,
//     loc = (64/66)*(2*t[3:6]-1)  (masks use c = 0).
// ---------------------------------------------------------------------------
__global__ void theta_params_kernel(const float* __restrict__ theta,
                                    float* __restrict__ params)
{
  const int b = threadIdx.x;
  if (b >= NB) return;
  const float* t = theta + b * 6;
  const float phi = t[0] * (2.f * PI_F) - PI_F;
  const float th  = t[1] * (2.f * PI_F) - PI_F;
  const float psi = t[2] * (2.f * PI_F) - PI_F;
  float cph, sph, cth, sth, cps, sps;
  sincosf(phi, &sph, &cph);
  sincosf(th,  &sth, &cth);
  sincosf(psi, &sps, &cps);
  const float A[9]  = { cph, -sph, 0.f,  sph, cph, 0.f,  0.f, 0.f, 1.f };
  const float Bm[9] = { cth, 0.f, sth,   0.f, 1.f, 0.f, -sth, 0.f, cth };
  const float Cm[9] = { cps, -sps, 0.f,  sps, cps, 0.f,  0.f, 0.f, 1.f };
  float AB[9], R[9], M[9];
  for (int i = 0; i < 3; ++i)
    for (int j = 0; j < 3; ++j) {
      float s = 0.f;
      for (int k = 0; k < 3; ++k) s += A[i*3+k] * Bm[k*3+j];
      AB[i*3+j] = s;
    }
  for (int i = 0; i < 3; ++i)
    for (int j = 0; j < 3; ++j) {
      float s = 0.f;
      for (int k = 0; k < 3; ++k) s += AB[i*3+k] * Cm[k*3+j];
      R[i*3+j] = s;
    }
  for (int i = 0; i < 3; ++i)
    for (int j = 0; j < 3; ++j) M[i*3+j] = R[j*3+i];   // transpose
  const float sc = 64.f / 66.f;
  const float loc[3] = { sc*(2.f*t[3]-1.f), sc*(2.f*t[4]-1.f), sc*(2.f*t[5]-1.f) };
  float* p = params + b * 16;
  #pragma unroll
  for (int i = 0; i < 9; ++i) p[i] = M[i];
  for (int i = 0; i < 3; ++i)
    p[9+i] = -(M[i*3+0]*loc[0] + M[i*3+1]*loc[1] + M[i*3+2]*loc[2]);
}

// ---------------------------------------------------------------------------
// Trilinear samplers (faithful to reference clip semantics)
// ---------------------------------------------------------------------------
__device__ __forceinline__ float sample_clamp64(const float* __restrict__ img,
                                                float x, float y, float z)
{
  const float MX = 63.f;
  float x0 = fminf(fmaxf(floorf(x), 0.f), MX), x1 = fminf(x0 + 1.f, MX);
  float y0 = fminf(fmaxf(floorf(y), 0.f), MX), y1 = fminf(y0 + 1.f, MX);
  float z0 = fminf(fmaxf(floorf(z), 0.f), MX), z1 = fminf(z0 + 1.f, MX);
  float dx = x - x0, dy = y - y0, dz = z - z0;
  int a0 = (int)x0, a1 = (int)x1, b0 = (int)y0, b1 = (int)y1, c0 = (int)z0, c1 = (int)z1;
  auto g = [&](int a, int b, int c) { return img[(a * 64 + b) * 64 + c]; };
  return (1-dx)*(1-dy)*(1-dz)*g(a0,b0,c0) + (1-dx)*(1-dy)*dz*g(a0,b0,c1)
       + (1-dx)*dy*(1-dz)*g(a0,b1,c0)     + (1-dx)*dy*dz*g(a0,b1,c1)
       + dx*(1-dy)*(1-dz)*g(a1,b0,c0)     + dx*(1-dy)*dz*g(a1,b0,c1)
       + dx*dy*(1-dz)*g(a1,b1,c0)         + dx*dy*dz*g(a1,b1,c1);
}

// sample the implicitly zero-padded 66^3 volume (border index 0/65 -> 0)
__device__ __forceinline__ float sample_pad66(const float* __restrict__ img,
                                              float x, float y, float z)
{
  const float MX = 65.f;
  float x0 = fminf(fmaxf(floorf(x), 0.f), MX), x1 = fminf(x0 + 1.f, MX);
  float y0 = fminf(fmaxf(floorf(y), 0.f), MX), y1 = fminf(y0 + 1.f, MX);
  float z0 = fminf(fmaxf(floorf(z), 0.f), MX), z1 = fminf(z0 + 1.f, MX);
  float dx = x - x0, dy = y - y0, dz = z - z0;
  int a0 = (int)x0, a1 = (int)x1, b0 = (int)y0, b1 = (int)y1, c0 = (int)z0, c1 = (int)z1;
  auto g = [&](int a, int b, int c) -> float {
    if (a < 1 || a > 64 || b < 1 || b > 64 || c < 1 || c > 64) return 0.f;
    return img[((a - 1) * 64 + (b - 1)) * 64 + (c - 1)];
  };
  return (1-dx)*(1-dy)*(1-dz)*g(a0,b0,c0) + (1-dx)*(1-dy)*dz*g(a0,b0,c1)
       + (1-dx)*dy*(1-dz)*g(a0,b1,c0)     + (1-dx)*dy*dz*g(a0,b1,c1)
       + dx*(1-dy)*(1-dz)*g(a1,b0,c0)     + dx*(1-dy)*dz*g(a1,b0,c1)
       + dx*dy*(1-dz)*g(a1,b1,c0)         + dx*dy*dz*g(a1,b1,c1);
}

// ---------------------------------------------------------------------------
// K2: all three warps, one thread per voxel.
// ---------------------------------------------------------------------------
__global__ __launch_bounds__(256) void warp_kernel(
    const float* __restrict__ X,  const float* __restrict__ m1,
    const float* __restrict__ m2, const float* __restrict__ params,
    float* __restrict__ Xt, float* __restrict__ M1t, float* __restrict__ M2t)
{
  __shared__ float P[12];
  const int b = blockIdx.x >> 10;                 // 1024 blocks per batch
  if (threadIdx.x < 12) P[threadIdx.x] = params[b * 16 + threadIdx.x];
  __syncthreads();
  const int v = ((blockIdx.x & 1023) << 8) + threadIdx.x;
  const int d = v >> 12, h = (v >> 6) & 63, w = v & 63;
  const int gi = b * NVOX + v;

  // masks: 64-point grid, zero translation
  {
    float gx = d * (2.f / 63.f) - 1.f;
    float gy = h * (2.f / 63.f) - 1.f;
    float gz = w * (2.f / 63.f) - 1.f;
    float tx = P[0]*gx + P[1]*gy + P[2]*gz;
    float ty = P[3]*gx + P[4]*gy + P[5]*gz;
    float tz = P[6]*gx + P[7]*gy + P[8]*gz;
    float x = (tx + 1.f) * 31.5f, y = (ty + 1.f) * 31.5f, z = (tz + 1.f) * 31.5f;
    M1t[gi] = sample_clamp64(m1 + (size_t)b * NVOX, x, y, z);
    M2t[gi] = sample_clamp64(m2 + (size_t)b * NVOX, x, y, z);
  }
  // X: 66-point grid at padded index (d+1,h+1,w+1), with translation
  {
    float gx = (d + 1) * (2.f / 65.f) - 1.f;
    float gy = (h + 1) * (2.f / 65.f) - 1.f;
    float gz = (w + 1) * (2.f / 65.f) - 1.f;
    float tx = P[0]*gx + P[1]*gy + P[2]*gz + P[9];
    float ty = P[3]*gx + P[4]*gy + P[5]*gz + P[10];
    float tz = P[6]*gx + P[7]*gy + P[8]*gz + P[11];
    float x = (tx + 1.f) * 32.5f, y = (ty + 1.f) * 32.5f, z = (tz + 1.f) * 32.5f;
    Xt[gi] = sample_pad66(X + (size_t)b * NVOX, x, y, z);
  }
}

// ---------------------------------------------------------------------------
// K3: forward shifted-DFT along W then H for one (b,d) plane of X_t or Y.
// ---------------------------------------------------------------------------
__global__ __launch_bounds__(256) void fft_plane_fwd(
    const float* __restrict__ Xt, const float* __restrict__ Y,
    const float* __restrict__ TFre, const float* __restrict__ TFim,
    float* __restrict__ FXre, float* __restrict__ FXim,
    float* __restrict__ FYre, float* __restrict__ FYim)
{
  __shared__ __align__(16) float Fre[4096], Fim[4096];  // shifted DFT matrix
  __shared__ __align__(16) float Pre[4096];             // real input plane
  __shared__ __align__(16) float Qre[4096], Qim[4096];  // after W pass
  const int tid = threadIdx.x;
  const int arr = blockIdx.x & 1;
  const int pb  = blockIdx.x >> 1;
  const int b = pb >> 6, d = pb & 63;
  const size_t base = (size_t)b * NVOX + (size_t)d * 4096;
  const float* src = (arr ? Y : Xt) + base;

  #pragma unroll
  for (int i = 0; i < 4; ++i) {
    const int q = (tid + 256 * i) * 4;                  // float4 granularity
    copy_g2lds_b128(&Fre[q], TFre + q);
    copy_g2lds_b128(&Fim[q], TFim + q);
    copy_g2lds_b128(&Pre[q], src + q);
  }
  async_join();

  const int wave = tid >> 5, lane = tid & 31;
  // pass along W: Q[r,k] = sum_j P[r,j] * F[k,j]   (B = F^T)
  #pragma unroll
  for (int t = 2 * wave; t < 2 * wave + 2; ++t) {
    const int tm = (t >> 2) << 4, tn = (t & 3) << 4;
    v8f re, im;
    cgemm_tile<true, false>(Pre + tm * 64, nullptr, 64,
                            Fre + tn * 64, Fim + tn * 64, 1, 64, lane, re, im);
    store_tile(Qre + tm * 64 + tn, 64, re, lane);
    store_tile(Qim + tm * 64 + tn, 64, im, lane);
  }
  __syncthreads();

  float* dre = (arr ? FYre : FXre) + base;
  float* dim = (arr ? FYim : FXim) + base;
  // pass along H: out[k2,k] = sum_r F[k2,r] * Q[r,k]
  #pragma unroll
  for (int t = 2 * wave; t < 2 * wave + 2; ++t) {
    const int tm = (t >> 2) << 4, tn = (t & 3) << 4;
    v8f re, im;
    cgemm_tile<false, false>(Fre + tm * 64, Fim + tm * 64, 64,
                             Qre + tn, Qim + tn, 64, 1, lane, re, im);
    store_tile(dre + tm * 64 + tn, 64, re, lane);
    store_tile(dim + tm * 64 + tn, 64, im, lane);
  }
}

// ---------------------------------------------------------------------------
// K4: along-D pass, fully fused: fwd-D(FX), fwd-D(FY), mask MAC, inv-D -> S1.
// One block per (b, 64-column chunk of the 4096 (h,w) columns).
// ---------------------------------------------------------------------------
__global__ __launch_bounds__(256) void dline_kernel(
    const float* __restrict__ FXre, const float* __restrict__ FXim,
    const float* __restrict__ FYre, const float* __restrict__ FYim,
    const float* __restrict__ M1t,  const float* __restrict__ M2t,
    const float* __restrict__ TFre, const float* __restrict__ TFim,
    const float* __restrict__ TIre, const float* __restrict__ TIim,
    float* __restrict__ S1re, float* __restrict__ S1im)
{
  __shared__ __align__(16) float Ffre[4096], Ffim[4096]; // shifted forward DFT
  __shared__ __align__(16) float Fire[4096], Fiim[4096]; // inverse DFT (1/64)
  __shared__ __align__(16) float Ure[4096], Uim[4096];   // FX chunk (rows = D)
  __shared__ __align__(16) float Vre[4096], Vim[4096];   // FY chunk
  __shared__ __align__(16) float Sre[4096], Sim[4096];   // combined spectrum
  const int tid = threadIdx.x;
  const int b  = blockIdx.x >> 6;
  const int cc = (blockIdx.x & 63) << 6;     // column base within (h,w)
  const size_t base = (size_t)b * NVOX + cc;

  #pragma unroll
  for (int i = 0; i < 4; ++i) {
    const int q  = tid + 256 * i;            // 1024 float4 per 16 KB array
    const int q4 = q * 4;
    copy_g2lds_b128(&Ffre[q4], TFre + q4);
    copy_g2lds_b128(&Ffim[q4], TFim + q4);
    copy_g2lds_b128(&Fire[q4], TIre + q4);
    copy_g2lds_b128(&Fiim[q4], TIim + q4);
    const int row = q >> 4, c4 = (q & 15) * 4;            // 16 float4 per row
    const size_t g = base + (size_t)row * 4096 + c4;
    copy_g2lds_b128(&Ure[q4], FXre + g);
    copy_g2lds_b128(&Uim[q4], FXim + g);
    copy_g2lds_b128(&Vre[q4], FYre + g);
    copy_g2lds_b128(&Vim[q4], FYim + g);
  }
  async_join();

  const int wave = tid >> 5, lane = tid & 31;
  const int half = (lane >> 4) & 1, lr = lane & 15;
  #pragma unroll
  for (int t = 2 * wave; t < 2 * wave + 2; ++t) {
    const int tm = (t >> 2) << 4, tn = (t & 3) << 4;
    v8f xr, xi, yr, yi;
    cgemm_tile<false, false>(Ffre + tm * 64, Ffim + tm * 64, 64,
                             Ure + tn, Uim + tn, 64, 1, lane, xr, xi);
    cgemm_tile<false, false>(Ffre + tm * 64, Ffim + tm * 64, 64,
                             Vre + tn, Vim + tn, 64, 1, lane, yr, yi);
    v8f sr, si;
    #pragma unroll
    for (int r = 0; r < 8; ++r) {
      const int row = tm + r + 8 * half;
      const size_t g = base + (size_t)row * 4096 + tn + lr;
      const float a = M1t[g], bm = M2t[g];
      sr[r] = xr[r] * a + yr[r] * bm;
      si[r] = xi[r] * a + yi[r] * bm;
    }
    store_tile(Sre + tm * 64 + tn, 64, sr, lane);
    store_tile(Sim + tm * 64 + tn, 64, si, lane);
  }
  __syncthreads();

  #pragma unroll
  for (int t = 2 * wave; t < 2 * wave + 2; ++t) {
    const int tm = (t >> 2) << 4, tn = (t & 3) << 4;
    v8f re, im;
    cgemm_tile<false, false>(Fire + tm * 64, Fiim + tm * 64, 64,
                             Sre + tn, Sim + tn, 64, 1, lane, re, im);
    store_tile(S1re + base + (size_t)tm * 4096 + tn, 4096, re, lane);
    store_tile(S1im + base + (size_t)tm * 4096 + tn, 4096, im, lane);
  }
}

// ---------------------------------------------------------------------------
// K5: inverse DFT along W then H for one (b,d) plane; write real part.
// ---------------------------------------------------------------------------
__global__ __launch_bounds__(256) void fft_plane_inv(
    const float* __restrict__ S1re, const float* __restrict__ S1im,
    const float* __restrict__ TIre, const float* __restrict__ TIim,
    float* __restrict__ out)
{
  __shared__ __align__(16) float Fire[4096], Fiim[4096];
  __shared__ __align__(16) float Pre[4096], Pim[4096];
  __shared__ __align__(16) float Qre[4096], Qim[4096];
  const int tid = threadIdx.x;
  const int b = blockIdx.x >> 6, d = blockIdx.x & 63;
  const size_t base = (size_t)b * NVOX + (size_t)d * 4096;

  #pragma unroll
  for (int i = 0; i < 4; ++i) {
    const int q = (tid + 256 * i) * 4;
    copy_g2lds_b128(&Fire[q], TIre + q);
    copy_g2lds_b128(&Fiim[q], TIim + q);
    copy_g2lds_b128(&Pre[q], S1re + base + q);
    copy_g2lds_b128(&Pim[q], S1im + base + q);
  }
  async_join();

  const int wave = tid >> 5, lane = tid & 31;
  // inverse along W: Q[r,w] = sum_k P[r,k] * Fi[w,k]   (B = Fi^T)
  #pragma unroll
  for (int t = 2 * wave; t < 2 * wave + 2; ++t) {
    const int tm = (t >> 2) << 4, tn = (t & 3) << 4;
    v8f re, im;
    cgemm_tile<false, false>(Pre + tm * 64, Pim + tm * 64, 64,
                             Fire + tn * 64, Fiim + tn * 64, 1, 64, lane, re, im);
    store_tile(Qre + tm * 64 + tn, 64, re, lane);
    store_tile(Qim + tm * 64 + tn, 64, im, lane);
  }
  __syncthreads();

  // inverse along H, real part only: out[h,w] = Re{ sum_k Fi[h,k] * Q[k,w] }
  #pragma unroll
  for (int t = 2 * wave; t < 2 * wave + 2; ++t) {
    const int tm = (t >> 2) << 4, tn = (t & 3) << 4;
    v8f re, im;
    cgemm_tile<false, true>(Fire + tm * 64, Fiim + tm * 64, 64,
                            Qre + tn, Qim + tn, 64, 1, lane, re, im);
    store_tile(out + base + (size_t)tm * 64 + tn, 64, re, lane);
  }
}

// ---------------------------------------------------------------------------
extern "C" void kernel_launch(void* const* d_in, const int* in_sizes, int n_in,
                              void* d_out, int out_size, void* d_ws, size_t ws_size,
                              hipStream_t stream)
{
  (void)in_sizes; (void)n_in; (void)out_size; (void)ws_size;
  const float* X     = (const float*)d_in[0];
  const float* Y     = (const float*)d_in[1];
  const float* m1    = (const float*)d_in[2];
  const float* m2    = (const float*)d_in[3];
  const float* theta = (const float*)d_in[4];

  float* out = (float*)d_out;            // [out | M1_t | M2_t]
  float* M1t = out + (size_t)NTOT;
  float* M2t = out + (size_t)2 * NTOT;

  float* ws     = (float*)d_ws;          // ~118 MB of f32 scratch
  float* params = ws;                    // 16 * 16
  float* Xt     = ws + 256;
  float* FXre   = Xt   + (size_t)NTOT;
  float* FXim   = FXre + (size_t)NTOT;
  float* FYre   = FXim + (size_t)NTOT;
  float* FYim   = FYre + (size_t)NTOT;
  float* S1re   = FYim + (size_t)NTOT;
  float* S1im   = S1re + (size_t)NTOT;
  float* TFre   = S1im + (size_t)NTOT;   // twiddle tables (4 x 16 KB)
  float* TFim   = TFre + 4096;
  float* TIre   = TFim + 4096;
  float* TIim   = TIre + 4096;

  twiddle_kernel<<<1, 256, 0, stream>>>(TFre, TFim, TIre, TIim);
  theta_params_kernel<<<1, 32, 0, stream>>>(theta, params);
  warp_kernel<<<NTOT / 256, 256, 0, stream>>>(X, m1, m2, params, Xt, M1t, M2t);
  fft_plane_fwd<<<2 * NB * 64, 256, 0, stream>>>(Xt, Y, TFre, TFim,
                                                 FXre, FXim, FYre, FYim);
  dline_kernel<<<NB * 64, 256, 0, stream>>>(FXre, FXim, FYre, FYim, M1t, M2t,
                                            TFre, TFim, TIre, TIim, S1re, S1im);
  fft_plane_inv<<<NB * 64, 256, 0, stream>>>(S1re, S1im, TIre, TIim, out);
}